// Weldon_12369505812883
// MI455X (gfx1250) — compile-verified
//
#include <hip/hip_runtime.h>
#include <math.h>

// WELDON pooling for MI455X (gfx1250, wave32).
// Phase 1: fp32 GEMV via V_WMMA_F32_16X16X4_F32 (memory-bound, ~56us floor @ 23.3 TB/s).
// Phase 2: exact top-100/bottom-100 via radix threshold search in LDS, deterministic sums.

typedef __attribute__((ext_vector_type(2))) float v2f;
typedef __attribute__((ext_vector_type(8))) float v8f;

#define BATCH 16
#define NPTS  10000
#define FDIM  2048
#define NTOP  100
#define NBOT  100
#define WAVES_PER_BLOCK 8

// ---------------- Phase 1: scores = x @ W + b via f32 WMMA ----------------
// One wave handles 16 consecutive n-indices (one 16-row A tile), chaining
// 16x16x4 f32 WMMAs over K=2048. B = W[k..k+3] broadcast to all 16 columns,
// so D[i][j] is the score of row i for every j.
__global__ __launch_bounds__(WAVES_PER_BLOCK * 32)
void weldon_scores_wmma(const float* __restrict__ x,
                        const float* __restrict__ W,
                        const float* __restrict__ bias,
                        float* __restrict__ scores) {
  const int lane  = threadIdx.x & 31;
  const int wid   = threadIdx.x >> 5;
  const int group = blockIdx.x * WAVES_PER_BLOCK + wid;
  const int GROUPS_PER_BATCH = NPTS / 16;               // 625
  if (group >= BATCH * GROUPS_PER_BATCH) return;        // wave-uniform exit
  const int bb = group / GROUPS_PER_BATCH;
  const int n0 = (group % GROUPS_PER_BATCH) * 16;

  // A 16x4 f32 layout: lanes 0-15 hold M=0..15 (K=0,1 in 2 VGPRs),
  // lanes 16-31 hold M=0..15 (K=2,3). B 4x16 mirrors the K mapping.
  const int m    = lane & 15;
  const int koff = (lane >> 4) * 2;
  const float* __restrict__ xrow = x + (size_t)(bb * NPTS + n0 + m) * FDIM + koff;
  const float* __restrict__ wrow = W + koff;

  v8f c = {};
#pragma unroll 4
  for (int k = 0; k < FDIM; k += 4) {
    v2f a = *(const v2f*)(xrow + k);   // x[row][k+koff], x[row][k+koff+1]
    v2f w = *(const v2f*)(wrow + k);   // W[k+koff], W[k+koff+1] (same for all cols)
    // 8 args: (neg_a, A, neg_b, B, c_mod, C, reuse_a, reuse_b)
    c = __builtin_amdgcn_wmma_f32_16x16x4_f32(false, a, false, w,
                                              (short)0, c, false, false);
  }

  // C/D layout: VGPR r, lanes 0-15 -> (M=r, N=lane); lanes 16-31 -> (M=r+8, N=lane-16).
  // Column N==0 lives in lanes 0 and 16; together they hold all 16 row scores.
  const float bv = bias[0];
  if (m == 0) {
    const int mbase = (lane >> 4) * 8;
    float* __restrict__ srow = scores + bb * NPTS + n0 + mbase;
#pragma unroll
    for (int r = 0; r < 8; ++r) srow[r] = c[r] + bv;
  }
}

// ---------------- Phase 2: exact top/bottom-k mean + sigmoid ----------------
__device__ __forceinline__ unsigned f2key(float f) {
  unsigned u = __float_as_uint(f);
  return (u & 0x80000000u) ? ~u : (u | 0x80000000u);  // monotonic: bigger float -> bigger key
}
__device__ __forceinline__ float key2f(unsigned u) {
  unsigned b = (u & 0x80000000u) ? (u ^ 0x80000000u) : ~u;
  return __uint_as_float(b);
}

__global__ __launch_bounds__(1024)
void weldon_select(const float* __restrict__ scores, float* __restrict__ out) {
  __shared__ unsigned keys[NPTS];      // 40 KB (WGP LDS = 320 KB)
  __shared__ int      s_cnt;
  __shared__ float    s_red[1024];
  const int tid = threadIdx.x;
  const int b   = blockIdx.x;
  const float* __restrict__ srow = scores + b * NPTS;

  for (int i = tid; i < NPTS; i += 1024) keys[i] = f2key(srow[i]);
  __syncthreads();

  // --- 100th-largest key: max t s.t. count(key >= t) >= NTOP ---
  unsigned ttop = 0u;
  for (int bit = 31; bit >= 0; --bit) {
    const unsigned cand = ttop | (1u << bit);
    if (tid == 0) s_cnt = 0;
    __syncthreads();
    int c = 0;
    for (int i = tid; i < NPTS; i += 1024) c += (keys[i] >= cand);
    if (c) atomicAdd(&s_cnt, c);                 // integer: deterministic
    __syncthreads();
    if (s_cnt >= NTOP) ttop = cand;
    __syncthreads();
  }

  // --- 100th-smallest key: min t s.t. count(key <= t) >= NBOT ---
  unsigned tbot = 0xFFFFFFFFu;
  for (int bit = 31; bit >= 0; --bit) {
    const unsigned cand = tbot & ~(1u << bit);
    if (tid == 0) s_cnt = 0;
    __syncthreads();
    int c = 0;
    for (int i = tid; i < NPTS; i += 1024) c += (keys[i] <= cand);
    if (c) atomicAdd(&s_cnt, c);
    __syncthreads();
    if (s_cnt >= NBOT) tbot = cand;
    __syncthreads();
  }

  // --- sum of strict-top + tie count (deterministic tree reduction) ---
  if (tid == 0) s_cnt = 0;
  __syncthreads();
  float ps = 0.0f; int cg = 0;
  for (int i = tid; i < NPTS; i += 1024) {
    const unsigned u = keys[i];
    if (u > ttop) { ps += key2f(u); ++cg; }
  }
  if (cg) atomicAdd(&s_cnt, cg);
  s_red[tid] = ps;
  __syncthreads();
  for (int s = 512; s > 0; s >>= 1) {
    if (tid < s) s_red[tid] += s_red[tid + s];
    __syncthreads();
  }
  const float sum_top_strict = s_red[0];
  const int   cnt_gt         = s_cnt;
  __syncthreads();

  // --- sum of strict-bottom + tie count ---
  if (tid == 0) s_cnt = 0;
  __syncthreads();
  ps = 0.0f; cg = 0;
  for (int i = tid; i < NPTS; i += 1024) {
    const unsigned u = keys[i];
    if (u < tbot) { ps += key2f(u); ++cg; }
  }
  if (cg) atomicAdd(&s_cnt, cg);
  s_red[tid] = ps;
  __syncthreads();
  for (int s = 512; s > 0; s >>= 1) {
    if (tid < s) s_red[tid] += s_red[tid + s];
    __syncthreads();
  }

  if (tid == 0) {
    const float st   = sum_top_strict + (float)(NTOP - cnt_gt) * key2f(ttop);
    const float sb   = s_red[0]       + (float)(NBOT - s_cnt)  * key2f(tbot);
    const float mean = (st + sb) / (float)(NTOP + NBOT);
    out[b] = 1.0f / (1.0f + expf(-mean));
  }
}

// ---------------- launch ----------------
extern "C" void kernel_launch(void* const* d_in, const int* in_sizes, int n_in,
                              void* d_out, int out_size, void* d_ws, size_t ws_size,
                              hipStream_t stream) {
  (void)in_sizes; (void)n_in; (void)out_size; (void)ws_size;
  const float* x    = (const float*)d_in[0];   // [16,10000,2048] fp32
  const float* W    = (const float*)d_in[1];   // [1,2048] fp32
  const float* bias = (const float*)d_in[2];   // [1] fp32
  float* out    = (float*)d_out;               // [16] fp32
  float* scores = (float*)d_ws;                // B*NPTS fp32 = 640 KB scratch

  const int groups = BATCH * (NPTS / 16);                       // 10000 wave tiles
  const int blocks = (groups + WAVES_PER_BLOCK - 1) / WAVES_PER_BLOCK;
  weldon_scores_wmma<<<blocks, WAVES_PER_BLOCK * 32, 0, stream>>>(x, W, bias, scores);
  weldon_select<<<BATCH, 1024, 0, stream>>>(scores, out);
}